// RIPointTransformer_48223892799871
// MI455X (gfx1250) — compile-verified
//
#include <hip/hip_runtime.h>
#include <math.h>
#include <stdint.h>

// ---------------------------------------------------------------------------
// RIPointTransformer on MI455X (gfx1250, wave32).
//  K1: fused KNN  — Gram tiles via V_WMMA_F32_16X16X4_F32 + register top-16
//      (unsorted buffer + worst tracking; no final sort: the attention tail
//       is permutation-invariant in the neighbor axis)
//  K2: fused per-point PPF -> MLP(WMMA) -> K/V proj(WMMA) -> attention in LDS
//      xg gather uses GLOBAL_LOAD_ASYNC_TO_LDS_B128; memory clobbers only at
//      the fence points (first load, final s_wait_asynccnt)
//  K3: out @ Wo   — WMMA tiled GEMM
//  K4: LayerNorm + residual + ReLU
// ---------------------------------------------------------------------------

typedef __attribute__((ext_vector_type(2))) float v2f;
typedef __attribute__((ext_vector_type(8))) float v8f;

#define N_PTS 8192
#define CH    64
#define KNB   16
#define EPSF  1e-9f

static __device__ inline v8f wmma_k4(v2f a, v2f b, v8f c) {
  // V_WMMA_F32_16X16X4_F32 : D = A(16x4) * B(4x16) + C(16x16)
  return __builtin_amdgcn_wmma_f32_16x16x4_f32(false, a, false, b, (short)0, c,
                                               false, false);
}

static __device__ inline float ang3(float ux, float uy, float uz,
                                    float vx, float vy, float vz) {
  float cx = uy * vz - uz * vy;
  float cy = uz * vx - ux * vz;
  float cz = ux * vy - uy * vx;
  float cn = sqrtf(cx * cx + cy * cy + cz * cz + EPSF);
  float dt = ux * vx + uy * vy + uz * vz;
  return atan2f(cn, dt);
}

// ---------------------------------------------------------------------------
// Kernel 1: KNN.  One wave owns a 16-query tile; loops over 512 candidate
// tiles, each tile = one f32 WMMA producing (sq_j - 2 p_i.p_j).  Row-constant
// sq_i is omitted (does not change per-row ordering).  Top-16 kept in an
// UNSORTED register buffer with worst-slot tracking; the scan loop is rolled
// so the (rare) accept body exists exactly once.  The selected SET is what
// matters downstream (attention is permutation-invariant over neighbors), so
// no final sort is performed.
// ---------------------------------------------------------------------------
__global__ __launch_bounds__(256) void knn_wmma_kernel(
    const float* __restrict__ p, int* __restrict__ knn_idx) {
  __shared__ float tile[8][256];
  const int lane = threadIdx.x & 31;
  const int wave = threadIdx.x >> 5;
  const int qtile = blockIdx.x * 8 + wave;
  const int l = lane & 15;
  const int half = lane >> 4;
  const int row = qtile * 16 + l;  // lanes 16-31 mirror rows 0-15

  // A fragment (constant over the whole scan): [px,py,pz,1]
  const float px = p[row * 3 + 0], py = p[row * 3 + 1], pz = p[row * 3 + 2];
  v2f a;
  a.x = half ? pz : px;
  a.y = half ? 1.0f : py;

  float bd[KNB];
  int bi[KNB];
#pragma unroll
  for (int s = 0; s < KNB; ++s) { bd[s] = 3.4e38f; bi[s] = 0; }
  float worstVal = 3.4e38f;
  int worstSlot = 0;

  for (int ct = 0; ct < N_PTS / 16; ++ct) {
    const int j = ct * 16 + l;
    const float cx = p[j * 3 + 0], cy = p[j * 3 + 1], cz = p[j * 3 + 2];
    const float sq = cx * cx + cy * cy + cz * cz;
    v2f b;
    b.x = half ? (-2.0f * cz) : (-2.0f * cx);
    b.y = half ? sq : (-2.0f * cy);

    v8f c = {0.f, 0.f, 0.f, 0.f, 0.f, 0.f, 0.f, 0.f};
    c = wmma_k4(a, b, c);
#pragma unroll
    for (int r = 0; r < 8; ++r) tile[wave][(r + half * 8) * 16 + l] = c[r];

    // scan the 16 candidate distances of my row (rolled: one accept body)
#pragma unroll 1
    for (int c2 = 0; c2 < 16; ++c2) {
      const float dv = tile[wave][l * 16 + c2];
      if (__builtin_expect(dv < worstVal, 0)) {
        const int jj = ct * 16 + c2;
        // replace the current worst slot (flat cndmask writes)
#pragma unroll
        for (int s = 0; s < KNB; ++s) {
          const bool hit = (s == worstSlot);
          bd[s] = hit ? dv : bd[s];
          bi[s] = hit ? jj : bi[s];
        }
        // rescan for the new worst (unrolled max-reduction)
        worstVal = bd[0];
        worstSlot = 0;
#pragma unroll
        for (int s = 1; s < KNB; ++s) {
          const bool gt = bd[s] > worstVal;
          worstVal = gt ? bd[s] : worstVal;
          worstSlot = gt ? s : worstSlot;
        }
      }
    }
  }

  if (lane < 16) {
#pragma unroll
    for (int s = 0; s < KNB; ++s) knn_idx[row * KNB + s] = bi[s];
  }
}

// ---------------------------------------------------------------------------
// Kernel 2: fused per-point pipeline.  One wave per point.
//   LDS slice per wave (floats): [0,64) ppf  [64,128) q  [128,192) attn
//   R0 @256 : h1 then gathered xg (16x64)
//   R1 @1280: pe   R2 @2304: k   R3 @3328: v
// ---------------------------------------------------------------------------
#define WPW 4            // waves per workgroup
#define PWF 4352         // floats per wave slice

__global__ __launch_bounds__(WPW * 32) void point_attn_kernel(
    const float* __restrict__ p, const float* __restrict__ x,
    const float* __restrict__ nrm, const float* __restrict__ Wq,
    const float* __restrict__ Wk, const float* __restrict__ Wv,
    const float* __restrict__ w1, const float* __restrict__ b1,
    const float* __restrict__ w2, const float* __restrict__ b2,
    const int* __restrict__ knn_idx, float* __restrict__ attnout) {
  extern __shared__ float smem[];
  const int lane = threadIdx.x & 31;
  const int wave = threadIdx.x >> 5;
  float* sw = smem + wave * PWF;
  float* sPPF = sw;
  float* sQ = sw + 64;
  float* sATT = sw + 128;
  float* R0 = sw + 256;
  float* R1 = sw + 1280;
  float* R2 = sw + 2304;
  float* R3 = sw + 3328;
  const int n = blockIdx.x * WPW + wave;
  const int l = lane & 15;
  const int half = lane >> 4;
  const int koff = half * 2;

  __builtin_prefetch(Wk, 0, 3);
  __builtin_prefetch(Wv, 0, 3);
  __builtin_prefetch(w2, 0, 3);

  // ---- q = x[n] @ Wq : 2 columns per lane -------------------------------
#pragma unroll
  for (int t = 0; t < 2; ++t) {
    const int c = lane + 32 * t;
    float s = 0.f;
#pragma unroll 8
    for (int j = 0; j < CH; ++j) s = fmaf(x[n * CH + j], Wq[j * CH + c], s);
    sQ[c] = s;
  }

  // ---- PPF features for the 16 neighbors (lanes 0-15) -------------------
  const float pcx = p[n * 3 + 0], pcy = p[n * 3 + 1], pcz = p[n * 3 + 2];
  const float ncx = nrm[n * 3 + 0], ncy = nrm[n * 3 + 1], ncz = nrm[n * 3 + 2];
  if (lane < 16) {
    const int j = knn_idx[n * KNB + lane];
    const float dx = p[j * 3 + 0] - pcx;
    const float dy = p[j * 3 + 1] - pcy;
    const float dz = p[j * 3 + 2] - pcz;
    const float njx = nrm[j * 3 + 0], njy = nrm[j * 3 + 1], njz = nrm[j * 3 + 2];
    sPPF[lane * 4 + 0] = ang3(ncx, ncy, ncz, dx, dy, dz);
    sPPF[lane * 4 + 1] = ang3(njx, njy, njz, dx, dy, dz);
    sPPF[lane * 4 + 2] = ang3(ncx, ncy, ncz, njx, njy, njz);
    sPPF[lane * 4 + 3] = sqrtf(dx * dx + dy * dy + dz * dz + EPSF);
  }

  // ---- h1 = relu(ppf @ w1 + b1)  (16x4 @ 4x64) : 4 WMMAs ----------------
  {
    v2f a;
    a.x = sPPF[l * 4 + koff + 0];
    a.y = sPPF[l * 4 + koff + 1];
#pragma unroll
    for (int ct = 0; ct < 4; ++ct) {
      v2f b;
      b.x = w1[(koff + 0) * CH + ct * 16 + l];
      b.y = w1[(koff + 1) * CH + ct * 16 + l];
      v8f c = {0.f, 0.f, 0.f, 0.f, 0.f, 0.f, 0.f, 0.f};
      c = wmma_k4(a, b, c);
#pragma unroll
      for (int r = 0; r < 8; ++r) {
        const int rr = r + half * 8;
        const int cc = ct * 16 + l;
        R0[rr * CH + cc] = fmaxf(c[r] + b1[cc], 0.f);
      }
    }
  }

  // ---- pe = h1 @ w2 + b2  (16x64 @ 64x64) : 16 WMMAs per col tile -------
#pragma unroll
  for (int ct = 0; ct < 4; ++ct) {
    v8f c = {0.f, 0.f, 0.f, 0.f, 0.f, 0.f, 0.f, 0.f};
    for (int k0 = 0; k0 < CH; k0 += 4) {
      v2f a, b;
      a.x = R0[l * CH + k0 + koff + 0];
      a.y = R0[l * CH + k0 + koff + 1];
      b.x = w2[(k0 + koff + 0) * CH + ct * 16 + l];
      b.y = w2[(k0 + koff + 1) * CH + ct * 16 + l];
      c = wmma_k4(a, b, c);
    }
#pragma unroll
    for (int r = 0; r < 8; ++r) {
      const int rr = r + half * 8;
      const int cc = ct * 16 + l;
      R1[rr * CH + cc] = c[r] + b2[cc];
    }
  }

  // ---- gather xg = x[idx] into R0 via async LDS-DMA ---------------------
  // 16 rows x 64 floats = 4 KB; 8 x GLOBAL_LOAD_ASYNC_TO_LDS_B128
  // (16 B per lane per op).  LDS byte address = low 32 bits of flat address
  // (LDS aperture truncation, ISA 10.2).  Volatile asms keep their mutual
  // order, so only two fence points carry a memory clobber: the FIRST load
  // (WAR vs the pe reads of R0 above) and the WAIT (RAW vs the reads below).
  {
    const uint32_t lds_base = (uint32_t)(uintptr_t)R0;
#pragma unroll
    for (int i = 0; i < 8; ++i) {
      const int e = i * 128 + lane * 4;  // float index into the 16x64 tile
      const int r = e >> 6, cc = e & 63;
      const float* src = x + (size_t)knn_idx[n * KNB + r] * CH + cc;
      const uint32_t dst = lds_base + (uint32_t)e * 4u;
      if (i == 0) {
        asm volatile("global_load_async_to_lds_b128 %0, %1, off"
                     :: "v"(dst), "v"(src)
                     : "memory");
      } else {
        asm volatile("global_load_async_to_lds_b128 %0, %1, off"
                     :: "v"(dst), "v"(src));
      }
    }
    asm volatile("s_wait_asynccnt 0x0" ::: "memory");
  }

  // ---- k = xg @ Wk + pe ; v = xg @ Wv + pe ------------------------------
#pragma unroll
  for (int ct = 0; ct < 4; ++ct) {
    v8f ck = {0.f, 0.f, 0.f, 0.f, 0.f, 0.f, 0.f, 0.f};
    v8f cv = {0.f, 0.f, 0.f, 0.f, 0.f, 0.f, 0.f, 0.f};
    for (int k0 = 0; k0 < CH; k0 += 4) {
      v2f a, bk, bv;
      a.x = R0[l * CH + k0 + koff + 0];
      a.y = R0[l * CH + k0 + koff + 1];
      bk.x = Wk[(k0 + koff + 0) * CH + ct * 16 + l];
      bk.y = Wk[(k0 + koff + 1) * CH + ct * 16 + l];
      bv.x = Wv[(k0 + koff + 0) * CH + ct * 16 + l];
      bv.y = Wv[(k0 + koff + 1) * CH + ct * 16 + l];
      ck = wmma_k4(a, bk, ck);
      cv = wmma_k4(a, bv, cv);
    }
#pragma unroll
    for (int r = 0; r < 8; ++r) {
      const int rr = r + half * 8;
      const int cc = ct * 16 + l;
      R2[rr * CH + cc] = ck[r] + R1[rr * CH + cc];
      R3[rr * CH + cc] = cv[r] + R1[rr * CH + cc];
    }
  }

  // ---- multi-head attention (H=4, DH=16, K=16) --------------------------
  // pair id = lane + 32t : h = id>>4, kk = id&15 ; each 16-lane half is one
  // softmax group, so xor-shuffles with offset<16 stay in-group.
  const float scale = 0.25f;  // 1/sqrt(16)
#pragma unroll
  for (int t = 0; t < 2; ++t) {
    const int id = lane + 32 * t;
    const int h = id >> 4, kk = id & 15;
    float s = 0.f;
#pragma unroll
    for (int d = 0; d < 16; ++d)
      s = fmaf(sQ[h * 16 + d], R2[kk * CH + h * 16 + d], s);
    s *= scale;
    float m = s;
#pragma unroll
    for (int o = 8; o; o >>= 1) m = fmaxf(m, __shfl_xor(m, o, 32));
    const float e = __expf(s - m);
    float sum = e;
#pragma unroll
    for (int o = 8; o; o >>= 1) sum += __shfl_xor(sum, o, 32);
    sATT[id] = e / sum;
  }
#pragma unroll
  for (int t = 0; t < 2; ++t) {
    const int id = lane + 32 * t;
    const int h = id >> 4, d = id & 15;
    float o = 0.f;
#pragma unroll
    for (int kk = 0; kk < 16; ++kk)
      o = fmaf(sATT[h * 16 + kk], R3[kk * CH + h * 16 + d], o);
    attnout[n * CH + id] = o;
  }
}

// ---------------------------------------------------------------------------
// Kernel 3: out2 = attnout(8192x64) @ Wo(64x64) — WMMA tiles, one wave/tile.
// ---------------------------------------------------------------------------
__global__ __launch_bounds__(256) void out_gemm_kernel(
    const float* __restrict__ A, const float* __restrict__ Wo,
    float* __restrict__ out2) {
  const int lane = threadIdx.x & 31;
  const int wave = threadIdx.x >> 5;
  const int gw = blockIdx.x * 8 + wave;
  const int mt = gw >> 2;
  const int ctile = gw & 3;
  const int l = lane & 15;
  const int half = lane >> 4;
  const int koff = half * 2;

  v8f c = {0.f, 0.f, 0.f, 0.f, 0.f, 0.f, 0.f, 0.f};
  for (int k0 = 0; k0 < CH; k0 += 4) {
    v2f a, b;
    a.x = A[(mt * 16 + l) * CH + k0 + koff + 0];
    a.y = A[(mt * 16 + l) * CH + k0 + koff + 1];
    b.x = Wo[(k0 + koff + 0) * CH + ctile * 16 + l];
    b.y = Wo[(k0 + koff + 1) * CH + ctile * 16 + l];
    c = wmma_k4(a, b, c);
  }
#pragma unroll
  for (int r = 0; r < 8; ++r)
    out2[(mt * 16 + r + half * 8) * CH + ctile * 16 + l] = c[r];
}

// ---------------------------------------------------------------------------
// Kernel 4: LayerNorm + residual + ReLU, one thread per row.
// ---------------------------------------------------------------------------
__global__ __launch_bounds__(256) void ln_res_relu_kernel(
    const float* __restrict__ out2, const float* __restrict__ x,
    const float* __restrict__ g, const float* __restrict__ b,
    float* __restrict__ out) {
  const int row = blockIdx.x * 256 + threadIdx.x;
  float mu = 0.f;
#pragma unroll 8
  for (int j = 0; j < CH; ++j) mu += out2[row * CH + j];
  mu *= (1.0f / CH);
  float var = 0.f;
#pragma unroll 8
  for (int j = 0; j < CH; ++j) {
    const float d = out2[row * CH + j] - mu;
    var = fmaf(d, d, var);
  }
  var *= (1.0f / CH);
  const float inv = rsqrtf(var + 1e-5f);
#pragma unroll 8
  for (int j = 0; j < CH; ++j) {
    const float v =
        (out2[row * CH + j] - mu) * inv * g[j] + b[j] + x[row * CH + j];
    out[row * CH + j] = fmaxf(v, 0.f);
  }
}

// ---------------------------------------------------------------------------
extern "C" void kernel_launch(void* const* d_in, const int* in_sizes, int n_in,
                              void* d_out, int out_size, void* d_ws,
                              size_t ws_size, hipStream_t stream) {
  const float* p   = (const float*)d_in[0];
  const float* x   = (const float*)d_in[1];
  const float* nrm = (const float*)d_in[2];
  const float* Wq  = (const float*)d_in[3];
  const float* Wk  = (const float*)d_in[4];
  const float* Wv  = (const float*)d_in[5];
  const float* Wo  = (const float*)d_in[6];
  const float* w1  = (const float*)d_in[7];
  const float* b1  = (const float*)d_in[8];
  const float* w2  = (const float*)d_in[9];
  const float* b2  = (const float*)d_in[10];
  const float* lng = (const float*)d_in[11];
  const float* lnb = (const float*)d_in[12];
  float* out = (float*)d_out;

  char* ws = (char*)d_ws;
  int* knn_idx   = (int*)ws;                                   // 512 KB
  float* attnout = (float*)(ws + (size_t)N_PTS * KNB * 4);     // 2 MB
  float* out2    = attnout + (size_t)N_PTS * CH;               // 2 MB

  knn_wmma_kernel<<<N_PTS / 16 / 8, 256, 0, stream>>>(p, knn_idx);

  point_attn_kernel<<<N_PTS / WPW, WPW * 32, WPW * PWF * sizeof(float),
                      stream>>>(p, x, nrm, Wq, Wk, Wv, w1, b1, w2, b2, knn_idx,
                                attnout);

  out_gemm_kernel<<<(N_PTS / 16) * 4 / 8, 256, 0, stream>>>(attnout, Wo, out2);

  ln_res_relu_kernel<<<N_PTS / 256, 256, 0, stream>>>(out2, x, lng, lnb, out);
}